// SubregionFeatureExtractor_51273319580279
// MI455X (gfx1250) — compile-verified
//
#include <hip/hip_runtime.h>
#include <hip/hip_bf16.h>

typedef float v2f __attribute__((ext_vector_type(2)));
typedef float v8f __attribute__((ext_vector_type(8)));

#define NB 2
#define NT 8
#define NCH 64
#define NVOX 110592                  // 48*48*48
#define NREG 32
#define NSEG 33                      // ids 0..32 (0 = background)
#define CHUNK 1024                   // voxels per wave
#define NCHUNK (NVOX / CHUNK)        // 108
#define CTILES (NCH / 16)            // 4
#define OUT_ELEMS (NB * NT * NREG * NCH)  // 32768

// ---------------------------------------------------------------- zero
__global__ void zero_kernel(float* __restrict__ out, float* __restrict__ counts) {
    int i = blockIdx.x * blockDim.x + threadIdx.x;
    if (i < OUT_ELEMS) out[i] = 0.0f;
    if (i < NB * NSEG) counts[i] = 0.0f;
}

// ---------------------------------------------------------------- counts
__global__ void count_kernel(const int* __restrict__ seg, float* __restrict__ counts) {
    __shared__ float hist[NB * NSEG];
    const int tid = threadIdx.x;
    for (int i = tid; i < NB * NSEG; i += blockDim.x) hist[i] = 0.0f;
    __syncthreads();
    const int total = NB * NVOX;
    for (int i = blockIdx.x * blockDim.x + tid; i < total; i += gridDim.x * blockDim.x) {
        int b = i / NVOX;
        int s = seg[i];
        atomicAdd(&hist[b * NSEG + s], 1.0f);   // ds_add_f32
    }
    __syncthreads();
    for (int i = tid; i < NB * NSEG; i += blockDim.x)
        if (hist[i] != 0.0f) atomicAdd(&counts[i], hist[i]);
}

// ---------------------------------------------------------------- WMMA segment-sum
// D[M=channel, N=region] += A[M=channel, K=voxel] * OneHot[K=voxel, N=region]
__global__ void __launch_bounds__(256)
seg_wmma_kernel(const float* __restrict__ feats, const int* __restrict__ seg,
                float* __restrict__ out) {
    const int wave = (blockIdx.x * blockDim.x + threadIdx.x) >> 5;
    const int lane = threadIdx.x & 31;

    int chunk = wave % NCHUNK;
    int tmp   = wave / NCHUNK;
    int ctile = tmp % CTILES; tmp /= CTILES;
    int t     = tmp % NT;
    int b     = tmp / NT;

    const int half = lane >> 4;   // K-group selector per ISA A/B layout
    const int m    = lane & 15;   // M for A (channel), N for B/D (region)

    // A-matrix 16x4 f32 layout: lane m holds row M=m; half 0 -> K={0,1}, half 1 -> K={2,3}
    const float* __restrict__ frow =
        feats + (size_t)((b * NT + t) * NCH + ctile * 16 + m) * NVOX;
    const int* __restrict__ srow = seg + (size_t)b * NVOX;
    const int n0 = chunk * CHUNK + 2 * half;

    v8f acc0 = {};   // regions 1..16  -> out rows 0..15
    v8f acc1 = {};   // regions 17..32 -> out rows 16..31
    const int rid0 = m + 1;
    const int rid1 = m + 17;

    for (int k = 0; k < CHUNK; k += 4) {
        v2f a = *(const v2f*)(frow + n0 + k);          // global_load_b64, streaming
        int2 s = *(const int2*)(srow + n0 + k);        // L2-resident (seg reused 32x)

        v2f b0, b1;                                    // one-hot B tiles (exact in f32)
        b0.x = (s.x == rid0) ? 1.0f : 0.0f;
        b0.y = (s.y == rid0) ? 1.0f : 0.0f;
        b1.x = (s.x == rid1) ? 1.0f : 0.0f;
        b1.y = (s.y == rid1) ? 1.0f : 0.0f;

        acc0 = __builtin_amdgcn_wmma_f32_16x16x4_f32(false, a, false, b0,
                                                     (short)0, acc0, false, false);
        acc1 = __builtin_amdgcn_wmma_f32_16x16x4_f32(false, a, false, b1,
                                                     (short)0, acc1, false, false);
    }

    // C/D 16x16 f32 layout: VGPR v, lane L -> M = v + 8*(L/16), N = L%16
    const int chbase = ctile * 16 + 8 * half;
    const size_t obase = (size_t)(b * NT + t) * NREG * NCH;
    for (int v = 0; v < 8; ++v) {
        int ch = chbase + v;
        atomicAdd(&out[obase + (size_t)m * NCH + ch],        acc0[v]);
        atomicAdd(&out[obase + (size_t)(m + 16) * NCH + ch], acc1[v]);
    }
}

// ---------------------------------------------------------------- mean finalize
__global__ void finalize_kernel(float* __restrict__ out, const float* __restrict__ counts) {
    int i = blockIdx.x * blockDim.x + threadIdx.x;
    if (i >= OUT_ELEMS) return;
    int r = (i >> 6) & 31;      // region index (0..31 -> id r+1)
    int b = i >> 14;            // / (NT*NREG*NCH) = / 16384
    float c = counts[b * NSEG + r + 1];
    out[i] = (c > 0.0f) ? out[i] / c : 0.0f;
}

// ---------------------------------------------------------------- launch
extern "C" void kernel_launch(void* const* d_in, const int* in_sizes, int n_in,
                              void* d_out, int out_size, void* d_ws, size_t ws_size,
                              hipStream_t stream) {
    const float* feats = (const float*)d_in[0];
    const int*   seg   = (const int*)d_in[1];
    float* out    = (float*)d_out;
    float* counts = (float*)d_ws;   // NB*NSEG floats

    zero_kernel<<<(OUT_ELEMS + 255) / 256, 256, 0, stream>>>(out, counts);
    count_kernel<<<128, 256, 0, stream>>>(seg, counts);

    const int total_waves = NB * NT * CTILES * NCHUNK;        // 6912
    const int blocks = total_waves * 32 / 256;                // 864 blocks of 8 waves
    seg_wmma_kernel<<<blocks, 256, 0, stream>>>(feats, seg, out);

    finalize_kernel<<<(OUT_ELEMS + 255) / 256, 256, 0, stream>>>(out, counts);
}